// SMLadaformer_25417616458207
// MI455X (gfx1250) — compile-verified
//
#include <hip/hip_runtime.h>
#include <math.h>

// Problem dims (fixed by the reference)
#define B_ 8
#define N_ 1024
#define D_ 1024
#define H_ 16
#define DEPTH_ 64
#define MLP_ 4096
#define EPS_ 1e-6f
#define INV_SQRT_DEPTH_ 0.125f   // 1/sqrt(64)

typedef float v2f __attribute__((ext_vector_type(2)));
typedef float v8f __attribute__((ext_vector_type(8)));

// ---------------------------------------------------------------------------
// fp32 WMMA GEMM, LDS-staged + double buffered.
//   C[M,N] = epilogue(A[M,K] @ B[K,N])
//   block = 256 threads = 8 waves, tile = 128(M) x 64(N), K-chunk = 16
//   grid.x = N/64, grid.y = M/128 ; all tiles full => EXEC all-ones for WMMA
//
// V_WMMA_F32_16X16X4_F32 fragments (ISA 7.12.2):
//   A (16x4): lanes 0-15 -> M=lane, v0=K0,v1=K1 ; lanes 16-31 -> K2,K3
//   B (4x16): lanes 0-15 -> N=lane, v0=K0,v1=K1 ; lanes 16-31 -> K2,K3
//   C/D     : vgpr i -> M=(lane>>4)*8+i, N=lane&15
//
// LDS: A tile row-major [128][16] padded to stride 20 floats; B tile stored
// TRANSPOSED [64 cols][16 k] stride 20, so a B fragment is one ds_load_b64.
// Stride 20 floats = 80B: 16B-aligned for b128 stores, 20*i mod 64 distinct
// for i=0..15 -> conflict-free fragment reads.
// ---------------------------------------------------------------------------
enum { EPI_BIAS = 0, EPI_BIAS_RES = 1, EPI_BIAS_GELU = 2, EPI_ACCUM = 3 };

template <int EPI>
__global__ __launch_bounds__(256) void wmma_gemm_f32(
    const float* __restrict__ A, int lda,
    const float* __restrict__ Bm, int ldb,
    const float* __restrict__ bias,     // per-column, unused for EPI_ACCUM
    const float* __restrict__ resid,    // same layout as C, EPI_BIAS_RES only
    float* __restrict__ C, int ldc, int K)
{
    constexpr int ST = 20;                       // padded LDS row stride (floats)
    __shared__ float Alds[2][128 * ST];          // 2 x 10240 B
    __shared__ float Blds[2][64 * ST];           // 2 x  5120 B

    const int tid  = threadIdx.x;
    const int lane = tid & 31;
    const int wv   = tid >> 5;                   // wave in block: 0..7
    const int hsel = lane >> 4;                  // 0/1 : K-pair / M-half select
    const int l16  = lane & 15;

    const int rowB = blockIdx.y * 128;
    const int col0 = blockIdx.x * 64;

    // cooperative staging maps
    const int arow = tid >> 2;                   // 0..63 (and +64)
    const int aseg = (tid & 3) * 4;              // 0,4,8,12 within k-chunk
    const int bkr  = tid >> 4;                   // 0..15 : k row
    const int bcg  = (tid & 15) * 4;             // 0..60 : col group

    const float* Ag0 = A + (size_t)(rowB + arow) * lda + aseg;
    const float* Ag1 = A + (size_t)(rowB + arow + 64) * lda + aseg;
    const float* Bg  = Bm + (size_t)bkr * ldb + col0 + bcg;

    const v8f vzero = {0.f, 0.f, 0.f, 0.f, 0.f, 0.f, 0.f, 0.f};
    v8f acc[4];
    acc[0] = vzero; acc[1] = vzero; acc[2] = vzero; acc[3] = vzero;

    // prefetch chunk 0 and stage into buffer 0
    float4 pa0 = *(const float4*)Ag0;
    float4 pa1 = *(const float4*)Ag1;
    float4 pb  = *(const float4*)Bg;
    *(float4*)(&Alds[0][arow * ST + aseg])        = pa0;
    *(float4*)(&Alds[0][(arow + 64) * ST + aseg]) = pa1;
    Blds[0][(bcg + 0) * ST + bkr] = pb.x;
    Blds[0][(bcg + 1) * ST + bkr] = pb.y;
    Blds[0][(bcg + 2) * ST + bkr] = pb.z;
    Blds[0][(bcg + 3) * ST + bkr] = pb.w;
    __syncthreads();

    const int NK = K >> 4;
    for (int kt = 0; kt < NK; ++kt) {
        const int cur = kt & 1;
        const int nxt = cur ^ 1;

        // issue global prefetch of next chunk before the matrix math
        if (kt + 1 < NK) {
            const int k0 = (kt + 1) << 4;
            pa0 = *(const float4*)(Ag0 + k0);
            pa1 = *(const float4*)(Ag1 + k0);
            pb  = *(const float4*)(Bg + (size_t)k0 * ldb);
        }

        // 16 WMMAs from LDS
        const float* Abase = &Alds[cur][(wv * 16 + l16) * ST + hsel * 2];
        const float* Bbase = &Blds[cur][l16 * ST + hsel * 2];
#pragma unroll
        for (int ks = 0; ks < 4; ++ks) {
            const v2f a = *(const v2f*)(Abase + ks * 4);
#pragma unroll
            for (int nt = 0; nt < 4; ++nt) {
                const v2f b = *(const v2f*)(Bbase + nt * 16 * ST + ks * 4);
                acc[nt] = __builtin_amdgcn_wmma_f32_16x16x4_f32(
                    false, a, false, b, (short)0, acc[nt], false, false);
            }
        }

        // stage next chunk into the other buffer
        if (kt + 1 < NK) {
            *(float4*)(&Alds[nxt][arow * ST + aseg])        = pa0;
            *(float4*)(&Alds[nxt][(arow + 64) * ST + aseg]) = pa1;
            Blds[nxt][(bcg + 0) * ST + bkr] = pb.x;
            Blds[nxt][(bcg + 1) * ST + bkr] = pb.y;
            Blds[nxt][(bcg + 2) * ST + bkr] = pb.z;
            Blds[nxt][(bcg + 3) * ST + bkr] = pb.w;
        }
        __syncthreads();
    }

    // epilogue
    const int rbase = rowB + wv * 16 + hsel * 8;
#pragma unroll
    for (int nt = 0; nt < 4; ++nt) {
        const int col = col0 + nt * 16 + l16;
        float bs = 0.f;
        if (EPI != EPI_ACCUM) bs = bias[col];
#pragma unroll
        for (int i = 0; i < 8; ++i) {
            const size_t off = (size_t)(rbase + i) * ldc + col;
            float v = acc[nt][i] + bs;
            if (EPI == EPI_BIAS_RES)  v += resid[off];
            if (EPI == EPI_BIAS_GELU) v = 0.5f * v * (1.f + erff(v * 0.70710678118f));
            if (EPI == EPI_ACCUM)     v = C[off] + acc[nt][i];
            C[off] = v;
        }
    }
}

// ---------------------------------------------------------------------------
// Small conditioning GEMM: C[M,N] = act(A[M,K] @ B[K,N] + bias)   (M == 8)
// ---------------------------------------------------------------------------
template <int RELU>
__global__ void small_gemm(const float* __restrict__ A, const float* __restrict__ Bm,
                           const float* __restrict__ bias, float* __restrict__ C,
                           int M, int Nn, int K)
{
    const int n = blockIdx.x * blockDim.x + threadIdx.x;
    const int m = blockIdx.y;
    if (n >= Nn || m >= M) return;
    const float* a = A + (size_t)m * K;
    float acc = 0.f;
    for (int k = 0; k < K; ++k) acc = fmaf(a[k], Bm[(size_t)k * Nn + n], acc);
    acc += bias[n];
    if (RELU) acc = fmaxf(acc, 0.f);
    C[(size_t)m * Nn + n] = acc;
}

// ---------------------------------------------------------------------------
// LayerNorm (no affine) + per-row modulation: y = xn * scale[b,n] + shift[b,n]
// one 256-thread block per row of length D_=1024 (float4 per thread)
// ---------------------------------------------------------------------------
__global__ __launch_bounds__(256) void ln_mod(const float* __restrict__ X,
                                              const float* __restrict__ scale,
                                              const float* __restrict__ shift,
                                              float* __restrict__ Y)
{
    __shared__ float s1[256];
    __shared__ float s2[256];
    const int row = blockIdx.x;                 // b*N + n
    const int tid = threadIdx.x;
    const float4 x = ((const float4*)(X + (size_t)row * D_))[tid];
    float sum = x.x + x.y + x.z + x.w;
    float sq  = x.x * x.x + x.y * x.y + x.z * x.z + x.w * x.w;
    s1[tid] = sum; s2[tid] = sq;
    __syncthreads();
    for (int st = 128; st > 0; st >>= 1) {
        if (tid < st) { s1[tid] += s1[tid + st]; s2[tid] += s2[tid + st]; }
        __syncthreads();
    }
    const float mean = s1[0] * (1.f / D_);
    const float var  = s2[0] * (1.f / D_) - mean * mean;
    const float rstd = rsqrtf(var + EPS_);
    const int b = row >> 10, n = row & (N_ - 1);
    const float sc = scale[b * D_ + n];
    const float sh = shift[b * D_ + n];
    float4 y;
    y.x = (x.x - mean) * rstd * sc + sh;
    y.y = (x.y - mean) * rstd * sc + sh;
    y.z = (x.z - mean) * rstd * sc + sh;
    y.w = (x.w - mean) * rstd * sc + sh;
    ((float4*)(Y + (size_t)row * D_))[tid] = y;
}

// ---------------------------------------------------------------------------
// attn logits: logits[b,h,n] = (q[b,n,:] . qa_w[:,h] + qa_b[h]) * 1/sqrt(d)
// one wave per (b,n) row; 8 waves per block
// ---------------------------------------------------------------------------
__global__ __launch_bounds__(256) void attn_logits(const float* __restrict__ Q,
                                                   const float* __restrict__ qa_w,
                                                   const float* __restrict__ qa_b,
                                                   float* __restrict__ logits)
{
    const int lane = threadIdx.x & 31;
    const int wv   = threadIdx.x >> 5;
    const int row  = blockIdx.x * 8 + wv;       // 0 .. B*N-1
    const int b = row >> 10, n = row & (N_ - 1);
    const float* q = Q + (size_t)row * D_;
    float acc[H_];
#pragma unroll
    for (int h = 0; h < H_; ++h) acc[h] = 0.f;
    for (int k0 = 0; k0 < D_; k0 += 32) {
        const float qv = q[k0 + lane];
        const float* wp = qa_w + (size_t)(k0 + lane) * H_;
#pragma unroll
        for (int h = 0; h < H_; ++h) acc[h] = fmaf(qv, wp[h], acc[h]);
    }
#pragma unroll
    for (int h = 0; h < H_; ++h) {
        float v = acc[h];
        for (int off = 16; off > 0; off >>= 1) v += __shfl_down(v, off, 32);
        if (lane == 0)
            logits[((size_t)(b * H_ + h)) * N_ + n] = (v + qa_b[h]) * INV_SQRT_DEPTH_;
    }
}

// softmax over N per (b,h); one block per row
__global__ __launch_bounds__(256) void softmax_n(const float* __restrict__ logits,
                                                 float* __restrict__ out)
{
    __shared__ float s[256];
    const int bh = blockIdx.x;
    const int tid = threadIdx.x;
    const float* in = logits + (size_t)bh * N_;
    float m = -INFINITY;
    for (int i = tid; i < N_; i += 256) m = fmaxf(m, in[i]);
    s[tid] = m; __syncthreads();
    for (int st = 128; st > 0; st >>= 1) {
        if (tid < st) s[tid] = fmaxf(s[tid], s[tid + st]);
        __syncthreads();
    }
    m = s[0];
    __syncthreads();
    float sum = 0.f;
    for (int i = tid; i < N_; i += 256) sum += expf(in[i] - m);
    s[tid] = sum; __syncthreads();
    for (int st = 128; st > 0; st >>= 1) {
        if (tid < st) s[tid] += s[tid + st];
        __syncthreads();
    }
    const float inv = 1.f / s[0];
    for (int i = tid; i < N_; i += 256)
        out[(size_t)bh * N_ + i] = expf(in[i] - m) * inv;
}

// global_q[b,h,d] = sum_n attn[b,h,n] * q[b,n,h*64+d]; one block per (b,h)
__global__ __launch_bounds__(256) void global_q_k(const float* __restrict__ attn,
                                                  const float* __restrict__ Q,
                                                  float* __restrict__ gq)
{
    __shared__ float s[256];
    const int bh = blockIdx.x;
    const int b = bh >> 4, h = bh & (H_ - 1);
    const int d = threadIdx.x & 63, grp = threadIdx.x >> 6;
    float acc = 0.f;
    for (int n = grp; n < N_; n += 4)
        acc = fmaf(attn[(size_t)bh * N_ + n],
                   Q[((size_t)(b * N_ + n)) * D_ + h * DEPTH_ + d], acc);
    s[threadIdx.x] = acc;
    __syncthreads();
    if (grp == 0)
        gq[(size_t)bh * DEPTH_ + d] = s[d] + s[64 + d] + s[128 + d] + s[192 + d];
}

// r[b,n,h*64+d] = gq[b,h,d] * k[...] * v[...]
__global__ void r_elem(const float* __restrict__ gq, const float* __restrict__ Kv,
                       const float* __restrict__ Vv, float* __restrict__ R)
{
    const size_t i = (size_t)blockIdx.x * blockDim.x + threadIdx.x;
    const int c = (int)(i & (D_ - 1));
    const size_t bn = i >> 10;
    const int b = (int)(bn >> 10);
    const int h = c >> 6, d = c & 63;
    R[i] = gq[((b * H_ + h) << 6) + d] * Kv[i] * Vv[i];
}

// out[r,c] = AO[r,c] + mlp_b2[c]   (seed for the accumulating MLP2 GEMMs)
__global__ void seed_out(const float* __restrict__ AO, const float* __restrict__ b2,
                         float* __restrict__ Out)
{
    const size_t i = (size_t)blockIdx.x * blockDim.x + threadIdx.x;
    Out[i] = AO[i] + b2[i & (D_ - 1)];
}

// ---------------------------------------------------------------------------
extern "C" void kernel_launch(void* const* d_in, const int* in_sizes, int n_in,
                              void* d_out, int out_size, void* d_ws, size_t ws_size,
                              hipStream_t stream)
{
    const float* inputs = (const float*)d_in[0];
    const float* z      = (const float*)d_in[1];
    const float* n1_hw  = (const float*)d_in[2];
    const float* n1_hb  = (const float*)d_in[3];
    const float* n1_gw  = (const float*)d_in[4];
    const float* n1_gb  = (const float*)d_in[5];
    const float* n1_bw  = (const float*)d_in[6];
    const float* n1_bb  = (const float*)d_in[7];
    const float* wq_w   = (const float*)d_in[8];
    const float* wq_b   = (const float*)d_in[9];
    const float* wk_w   = (const float*)d_in[10];
    const float* wk_b   = (const float*)d_in[11];
    const float* wv_w   = (const float*)d_in[12];
    const float* wv_b   = (const float*)d_in[13];
    const float* qa_w   = (const float*)d_in[14];
    const float* qa_b   = (const float*)d_in[15];
    const float* out_w  = (const float*)d_in[16];
    const float* out_b  = (const float*)d_in[17];
    const float* n2_hw  = (const float*)d_in[18];
    const float* n2_hb  = (const float*)d_in[19];
    const float* n2_gw  = (const float*)d_in[20];
    const float* n2_gb  = (const float*)d_in[21];
    const float* n2_bw  = (const float*)d_in[22];
    const float* n2_bb  = (const float*)d_in[23];
    const float* mlp_w1 = (const float*)d_in[24];
    const float* mlp_b1 = (const float*)d_in[25];
    const float* mlp_w2 = (const float*)d_in[26];
    const float* mlp_b2 = (const float*)d_in[27];

    float* out       = (float*)d_out;                       // [B,N,D]
    float* attn_maps = out + (size_t)B_ * N_ * D_;          // [B,H,N]

    // workspace: 4 x 32MB activation buffers + ~1MB small tensors (~130MB)
    const size_t T = (size_t)B_ * N_ * D_;                  // 8M floats
    float* ws = (float*)d_ws;
    float* X1 = ws;              // x_norm1 -> r
    float* Qb = ws + T;          // q -> attn_out
    float* Kb = ws + 2 * T;      // k -> x_norm2
    float* Vb = ws + 3 * T;      // v -> gelu chunk
    float* sm = ws + 4 * T;
    float* h1     = sm;
    float* scale1 = sm + 8192;
    float* shift1 = sm + 16384;
    float* h2     = sm + 24576;
    float* scale2 = sm + 32768;
    float* shift2 = sm + 40960;
    float* logits = sm + 49152;                 // [B*H*N] = 131072
    float* gq     = sm + 49152 + (size_t)B_ * H_ * N_;

    const dim3 blk(256);
    const dim3 gsmall(D_ / 256, B_);
    const dim3 ggemm(D_ / 64, (B_ * N_) / 128);             // (16, 64)
    const int  rows = B_ * N_;                              // 8192

    // conditioning for norm1
    small_gemm<1><<<gsmall, blk, 0, stream>>>(z,  n1_hw, n1_hb, h1,     B_, D_, D_);
    small_gemm<0><<<gsmall, blk, 0, stream>>>(h1, n1_gw, n1_gb, scale1, B_, D_, D_);
    small_gemm<0><<<gsmall, blk, 0, stream>>>(h1, n1_bw, n1_bb, shift1, B_, D_, D_);

    // x_norm1
    ln_mod<<<rows, blk, 0, stream>>>(inputs, scale1, shift1, X1);

    // q/k/v projections (fp32 WMMA)
    wmma_gemm_f32<EPI_BIAS><<<ggemm, blk, 0, stream>>>(X1, D_, wq_w, D_, wq_b, nullptr, Qb, D_, D_);
    wmma_gemm_f32<EPI_BIAS><<<ggemm, blk, 0, stream>>>(X1, D_, wk_w, D_, wk_b, nullptr, Kb, D_, D_);
    wmma_gemm_f32<EPI_BIAS><<<ggemm, blk, 0, stream>>>(X1, D_, wv_w, D_, wv_b, nullptr, Vb, D_, D_);

    // additive attention
    attn_logits<<<rows / 8, blk, 0, stream>>>(Qb, qa_w, qa_b, logits);
    softmax_n<<<B_ * H_, blk, 0, stream>>>(logits, attn_maps);
    global_q_k<<<B_ * H_, blk, 0, stream>>>(attn_maps, Qb, gq);
    r_elem<<<(unsigned)(T / 256), blk, 0, stream>>>(gq, Kb, Vb, X1);

    // attn_out = inputs + r @ out_w + out_b  -> Qb
    wmma_gemm_f32<EPI_BIAS_RES><<<ggemm, blk, 0, stream>>>(X1, D_, out_w, D_, out_b, inputs, Qb, D_, D_);

    // conditioning for norm2 + x_norm2 -> Kb
    small_gemm<1><<<gsmall, blk, 0, stream>>>(z,  n2_hw, n2_hb, h2,     B_, D_, D_);
    small_gemm<0><<<gsmall, blk, 0, stream>>>(h2, n2_gw, n2_gb, scale2, B_, D_, D_);
    small_gemm<0><<<gsmall, blk, 0, stream>>>(h2, n2_bw, n2_bb, shift2, B_, D_, D_);
    ln_mod<<<rows, blk, 0, stream>>>(Qb, scale2, shift2, Kb);

    // output = attn_out + gelu(x2 @ W1 + b1) @ W2 + b2, chunked over MLP dim
    seed_out<<<(unsigned)(T / 256), blk, 0, stream>>>(Qb, mlp_b2, out);
    for (int j = 0; j < 4; ++j) {
        const int j0 = j * 1024;
        wmma_gemm_f32<EPI_BIAS_GELU><<<ggemm, blk, 0, stream>>>(
            Kb, D_, mlp_w1 + j0, MLP_, mlp_b1 + j0, nullptr, Vb, 1024, D_);
        wmma_gemm_f32<EPI_ACCUM><<<ggemm, blk, 0, stream>>>(
            Vb, 1024, mlp_w2 + (size_t)j0 * D_, D_, nullptr, nullptr, out, D_, 1024);
    }
}